// CropAndResize_84095459656041
// MI455X (gfx1250) — compile-verified
//
#include <hip/hip_runtime.h>

#define CROP_H 14
#define CROP_W 14
#define NPOS (CROP_H * CROP_W)   // 196

namespace {
constexpr int kC = 256;
constexpr int kH = 200;
constexpr int kW = 200;
constexpr int kPlane = kH * kW;          // 40000
constexpr int kThreads = 256;
constexpr int kItems = kC * NPOS;        // 50176 = 196 * 256 exactly
}

__global__ __launch_bounds__(kThreads) void crop_resize_kernel(
    const float* __restrict__ image,
    const float* __restrict__ boxes,
    const int*   __restrict__ box_ind,
    float* __restrict__ out)
{
    // Per-position interpolation state, packed to one 16B record so the hot
    // loop needs a single ds_load_b128 per output element.
    //   x = top-left offset within channel plane
    //   y = dx(bit0) | dyw(bits 10:1) | valid(bit31)
    //   z = lx (bits), w = ly (bits)
    __shared__ int4 s_pos[NPOS];

    const int n   = blockIdx.x;
    const int tid = threadIdx.x;

    if (tid < NPOS) {
        const float y1 = boxes[n * 4 + 0];
        const float x1 = boxes[n * 4 + 1];
        const float y2 = boxes[n * 4 + 2];
        const float x2 = boxes[n * 4 + 3];
        const float hsc = (y2 - y1) * (float)(kH - 1) * (1.0f / (CROP_H - 1));
        const float wsc = (x2 - x1) * (float)(kW - 1) * (1.0f / (CROP_W - 1));

        const int iy = tid / CROP_W;
        const int ix = tid - iy * CROP_W;
        const float in_y = y1 * (float)(kH - 1) + (float)iy * hsc;
        const float in_x = x1 * (float)(kW - 1) + (float)ix * wsc;

        const bool vy = (in_y >= 0.0f) && (in_y <= (float)(kH - 1));
        const bool vx = (in_x >= 0.0f) && (in_x <= (float)(kW - 1));

        const float top  = floorf(in_y);
        const float left = floorf(in_x);

        int ti = (int)top;
        ti = ti < 0 ? 0 : (ti > kH - 1 ? kH - 1 : ti);
        int bi = ti + 1;
        bi = bi > kH - 1 ? kH - 1 : bi;
        int li = (int)left;
        li = li < 0 ? 0 : (li > kW - 1 ? kW - 1 : li);
        int ri = li + 1;
        ri = ri > kW - 1 ? kW - 1 : ri;

        const int dx  = ri - li;            // 0 or 1
        const int dyw = (bi - ti) * kW;     // 0 or W
        int packed = dx | (dyw << 1);
        if (vy && vx) packed |= (int)0x80000000u;

        int4 rec;
        rec.x = ti * kW + li;               // always in [0, kPlane)
        rec.y = packed;
        rec.z = __float_as_int(in_x - left);
        rec.w = __float_as_int(in_y - top);
        s_pos[tid] = rec;
    }
    __syncthreads();

    const int b = box_ind[n];
    const float* __restrict__ base = image + (size_t)b * (kC * kPlane);
    float* __restrict__ outn = out + (size_t)n * kItems;

    // Thread tid covers flat indices j = tid + k*256; j = c*196 + pos maps
    // straight onto the contiguous (c, iy, ix) output layout -> coalesced
    // NT stores. Consecutive lanes share a channel plane -> clustered gathers.
    int c   = tid / NPOS;
    int pos = tid - c * NPOS;
    int j   = tid;

    #pragma unroll 4
    for (int k = 0; k < kItems / kThreads; ++k) {   // 196 iterations
        const int4 rec = s_pos[pos];                 // ds_load_b128
        const int   tl     = rec.x;
        const int   packed = rec.y;
        const float lx     = __int_as_float(rec.z);
        const float ly     = __int_as_float(rec.w);
        const int dx  = packed & 1;
        const int dyw = (packed >> 1) & 0x3ff;

        // Clamped offsets are always in-plane: execute gathers unconditionally
        // (no EXEC divergence, loads can clause), select extrapolation at end.
        const float* __restrict__ p = base + c * kPlane + tl;
        const float tlv = p[0];
        const float trv = p[dx];
        const float blv = p[dyw];
        const float brv = p[dyw + dx];

        const float tv = fmaf(trv - tlv, lx, tlv);
        const float bv = fmaf(brv - blv, lx, blv);
        float v = fmaf(bv - tv, ly, tv);
        v = (packed < 0) ? v : 0.0f;                 // valid ? v : EXTRAPOLATION

        // Streamed 200MB output: non-temporal store keeps L2 for image reuse.
        __builtin_nontemporal_store(v, outn + j);

        j += kThreads;
        c += 1;
        pos += kThreads - NPOS;            // +60
        if (pos >= NPOS) { pos -= NPOS; c += 1; }
    }
}

extern "C" void kernel_launch(void* const* d_in, const int* in_sizes, int n_in,
                              void* d_out, int out_size, void* d_ws, size_t ws_size,
                              hipStream_t stream) {
    const float* image   = (const float*)d_in[0];
    const float* boxes   = (const float*)d_in[1];
    const int*   box_ind = (const int*)d_in[2];
    float*       out     = (float*)d_out;

    const int N = in_sizes[2];   // number of boxes (box_ind count)
    if (N <= 0) return;

    crop_resize_kernel<<<N, kThreads, 0, stream>>>(image, boxes, box_ind, out);
}